// GraphSAGE_Net_38156489457766
// MI455X (gfx1250) — compile-verified
//
#include <hip/hip_runtime.h>
#include <hip/hip_bf16.h>

#define N_NODES 50000
#define D_FEAT  128
#define HIDDEN  256
#define CLASSES 40
#define N_EDGES 800000
#define NCOL2P  48            // layer-2 columns padded to 3 n-tiles

typedef __attribute__((ext_vector_type(2))) float v2f;
typedef __attribute__((ext_vector_type(8))) float v8f;

// ---------------------------------------------------------------- zero scratch
__global__ void zero_kernel(float* __restrict__ p, long n) {
    long i = (long)blockIdx.x * blockDim.x + threadIdx.x;
    long stride = (long)gridDim.x * blockDim.x;
    for (; i < n; i += stride) p[i] = 0.0f;
}

// ---------------------------------------------------------------- in-degree
__global__ void count_kernel(const long long* __restrict__ dst,
                             float* __restrict__ counts, int E) {
    int e = blockIdx.x * blockDim.x + threadIdx.x;
    if (e < E) atomicAdd(&counts[(int)dst[e]], 1.0f);
}

// ------------------------------------------------- edge-parallel scatter-add
// one thread = one edge x one float4 feature chunk; chunks = 1 << cshift
__global__ void scatter_kernel(const float* __restrict__ feat,
                               const long long* __restrict__ src,
                               const long long* __restrict__ dst,
                               float* __restrict__ agg,
                               long total, int cshift, int D) {
    long t = (long)blockIdx.x * blockDim.x + threadIdx.x;
    if (t >= total) return;
    int e = (int)(t >> cshift);
    int c = ((int)t & ((1 << cshift) - 1)) << 2;   // feature offset (x4 floats)
    int s = (int)src[e];
    int d = (int)dst[e];
    const float4 v = *(const float4*)(feat + (long)s * D + c);
    float* o = agg + (long)d * D + c;
    atomicAdd(o + 0, v.x);
    atomicAdd(o + 1, v.y);
    atomicAdd(o + 2, v.z);
    atomicAdd(o + 3, v.w);
}

// ------------------------------------------- pack [Wl|Wr] (+ zero-pad layer 2)
__global__ void pack_weights_kernel(const float* __restrict__ Wl1,
                                    const float* __restrict__ Wr1,
                                    const float* __restrict__ Wl2,
                                    const float* __restrict__ Wr2,
                                    const float* __restrict__ bl2,
                                    float* __restrict__ Wcat1,
                                    float* __restrict__ Wcat2,
                                    float* __restrict__ bcat2) {
    int i = blockIdx.x * blockDim.x + threadIdx.x;
    if (i < HIDDEN * 2 * D_FEAT) {                       // 256 x 256
        int n = i / (2 * D_FEAT), k = i - n * (2 * D_FEAT);
        Wcat1[i] = (k < D_FEAT) ? Wl1[n * D_FEAT + k]
                                : Wr1[n * D_FEAT + (k - D_FEAT)];
    }
    if (i < NCOL2P * 2 * HIDDEN) {                       // 48 x 512 (rows 40+ = 0)
        int n = i / (2 * HIDDEN), k = i - n * (2 * HIDDEN);
        float v = 0.0f;
        if (n < CLASSES)
            v = (k < HIDDEN) ? Wl2[n * HIDDEN + k]
                             : Wr2[n * HIDDEN + (k - HIDDEN)];
        Wcat2[i] = v;
    }
    if (i < NCOL2P) bcat2[i] = (i < CLASSES) ? bl2[i] : 0.0f;
}

// ------------------------------------------------------- fused SAGE-conv GEMM
// out[m,n] = RELU?( [agg[m]/max(cnt,1) | xroot[m]] . Wcat[n] + bias[n] )
// A' tile staged in LDS (mean-scale folded in); B from pre-packed/padded Wcat
// (L2 resident).  WMMA k-loop: ds_load_b64 + global_load_b64 + v_wmma only.
// Epilogue: one uniform row-guard; full row-tiles do all
// 8 stores unguarded, the partial last tile guards each row.
template<int TWOK, int NTILES, int NSTORE, bool RELU>
__global__ void __launch_bounds__(256)
sage_gemm_wmma(const float* __restrict__ agg,
               const float* __restrict__ counts,
               const float* __restrict__ xroot,
               const float* __restrict__ Wcat,   // [NTILES*16, TWOK]
               const float* __restrict__ bias,   // [NTILES*16]
               float* __restrict__ out,          // [M, NSTORE]
               int M) {
    constexpr int KP  = TWOK / 2;
    constexpr int LDA = TWOK + 4;        // 16 lanes -> 16 distinct LDS banks
    __shared__ float lA[16 * LDA];

    const int m0  = blockIdx.x * 16;
    const int tid = threadIdx.x;

    // stage A' = [agg/deg | x] into LDS
    for (int i = tid; i < 16 * TWOK; i += 256) {
        int r = i / TWOK, c = i - r * TWOK;
        int row = m0 + r;
        float v = 0.0f;
        if (row < M) {
            v = (c < KP)
                ? agg[(long)row * KP + c] * (1.0f / fmaxf(counts[row], 1.0f))
                : xroot[(long)row * KP + (c - KP)];
        }
        lA[r * LDA + c] = v;
    }
    __syncthreads();

    const int wave = tid >> 5;
    const int lane = tid & 31;
    const int ln   = lane & 15;          // A: M index; B/C/D: N index
    const int hi   = lane >> 4;          // K half-select (A/B); +8 rows (C/D)
    const bool mok = (m0 + 16) <= M;     // uniform: whole row-tile in range

    for (int nt = wave; nt < NTILES; nt += 8) {
        const int ncol = nt * 16 + ln;

        const float* arow = &lA[ln * LDA + 2 * hi];
        const float* brow = Wcat + (long)ncol * TWOK + 2 * hi;

        v8f acc = {};
        #pragma unroll 8
        for (int k = 0; k < TWOK; k += 4) {
            v2f a = *(const v2f*)(arow + k);
            v2f b = *(const v2f*)(brow + k);
            acc = __builtin_amdgcn_wmma_f32_16x16x4_f32(
                      false, a, false, b, (short)0, acc, false, false);
        }

        const float bi = bias[ncol];
        float* orow = out + (long)m0 * NSTORE + ncol + (long)(8 * hi) * NSTORE;

        // C/D layout: VGPR v, lane l -> M = v + 8*(l>>4), N = l&15
        if (mok) {
            #pragma unroll
            for (int v = 0; v < 8; ++v) {
                float val = acc[v] + bi;
                if (RELU) val = fmaxf(val, 0.0f);
                orow[(long)v * NSTORE] = val;
            }
        } else {  // partial last row-tile (cold path)
            #pragma unroll
            for (int v = 0; v < 8; ++v) {
                int row = m0 + v + 8 * hi;
                if (row < M) {
                    float val = acc[v] + bi;
                    if (RELU) val = fmaxf(val, 0.0f);
                    out[(long)row * NSTORE + ncol] = val;
                }
            }
        }
    }
}

// ----------------------------------------- log_softmax (padded in, compact out)
__global__ void logsoftmax_kernel(const float* __restrict__ inp,  // [M, CIN]
                                  float* __restrict__ outp,       // [M, C]
                                  int M, int CIN, int C) {
    int row = blockIdx.x * blockDim.x + threadIdx.x;
    if (row >= M) return;
    const float* p = inp + (long)row * CIN;
    float* q = outp + (long)row * C;
    float m = -INFINITY;
    for (int j = 0; j < C; ++j) m = fmaxf(m, p[j]);
    float s = 0.0f;
    for (int j = 0; j < C; ++j) s += expf(p[j] - m);
    float ls = logf(s);
    for (int j = 0; j < C; ++j) q[j] = p[j] - m - ls;
}

extern "C" void kernel_launch(void* const* d_in, const int* in_sizes, int n_in,
                              void* d_out, int out_size, void* d_ws, size_t ws_size,
                              hipStream_t stream) {
    const float*     x   = (const float*)d_in[0];
    const long long* ei  = (const long long*)d_in[1];   // int64 edge_index [2,E]
    const float*     Wl1 = (const float*)d_in[2];
    const float*     bl1 = (const float*)d_in[3];
    const float*     Wr1 = (const float*)d_in[4];
    const float*     Wl2 = (const float*)d_in[5];
    const float*     bl2 = (const float*)d_in[6];
    const float*     Wr2 = (const float*)d_in[7];
    float* out = (float*)d_out;

    const long long* src = ei;
    const long long* dst = ei + N_EDGES;

    // workspace layout (floats):
    // [counts N][agg1 N*128][agg2 N*256][h N*256][out2p N*48][Wcat1][Wcat2][bcat2]
    float* ws     = (float*)d_ws;
    float* counts = ws;
    float* agg1   = counts + N_NODES;
    float* agg2   = agg1 + (long)N_NODES * D_FEAT;
    float* h      = agg2 + (long)N_NODES * HIDDEN;
    float* out2p  = h + (long)N_NODES * HIDDEN;          // padded [N, 48] logits
    float* Wcat1  = out2p + (long)N_NODES * NCOL2P;
    float* Wcat2  = Wcat1 + (long)HIDDEN * 2 * D_FEAT;
    float* bcat2  = Wcat2 + (long)NCOL2P * 2 * HIDDEN;

    // zero counts + agg1 + agg2 (contiguous)
    long nz = (long)N_NODES * (1 + D_FEAT + HIDDEN);
    zero_kernel<<<4096, 256, 0, stream>>>(ws, nz);

    // pack [Wl|Wr] weight matrices (layer-2 padded to 48 rows)
    pack_weights_kernel<<<(HIDDEN * 2 * D_FEAT + 255) / 256, 256, 0, stream>>>(
        Wl1, Wr1, Wl2, Wr2, bl2, Wcat1, Wcat2, bcat2);

    count_kernel<<<(N_EDGES + 255) / 256, 256, 0, stream>>>(dst, counts, N_EDGES);

    // layer 1: scatter-add x[src] -> agg1[dst]   (128 feats = 32 chunks/edge)
    long t1 = (long)N_EDGES * (D_FEAT / 4);
    scatter_kernel<<<(int)((t1 + 255) / 256), 256, 0, stream>>>(
        x, src, dst, agg1, t1, 5, D_FEAT);

    // h = relu([agg1/deg | x] @ Wcat1^T + bl1)   (K=256, 16 n-tiles, unguarded)
    sage_gemm_wmma<2 * D_FEAT, HIDDEN / 16, HIDDEN, true>
        <<<(N_NODES + 15) / 16, 256, 0, stream>>>(
            agg1, counts, x, Wcat1, bl1, h, N_NODES);

    // layer 2: scatter-add h[src] -> agg2[dst]   (256 feats = 64 chunks/edge)
    long t2 = (long)N_EDGES * (HIDDEN / 4);
    scatter_kernel<<<(int)((t2 + 255) / 256), 256, 0, stream>>>(
        h, src, dst, agg2, t2, 6, HIDDEN);

    // padded logits [N,48] = [agg2/deg | h] @ Wcat2^T + bcat2  (K=512, 3 tiles)
    sage_gemm_wmma<2 * HIDDEN, NCOL2P / 16, NCOL2P, false>
        <<<(N_NODES + 15) / 16, 256, 0, stream>>>(
            agg2, counts, h, Wcat2, bcat2, out2p, N_NODES);

    // log_softmax over the 40 real columns; compact write to d_out [N,40]
    logsoftmax_kernel<<<(N_NODES + 255) / 256, 256, 0, stream>>>(
        out2p, out, N_NODES, NCOL2P, CLASSES);
}